// HGT_28432683499970
// MI455X (gfx1250) — compile-verified
//
#include <hip/hip_runtime.h>
#include <math.h>

#define HIDC 160

typedef float v2f __attribute__((ext_vector_type(2)));
typedef float v8f __attribute__((ext_vector_type(8)));

// ---------------- WMMA f32 GEMM: C[M,N] = act(A[M,K] @ B[K,N] + bias) ----------------
// Block tile 64x32, 8 waves (4x2), each wave owns a 16x16 tile fed by
// V_WMMA_F32_16X16X4_F32. Double-buffered LDS staging: next K-tile is loaded
// into registers (unrolled -> one global_load clause, many loads in flight)
// while the current tile is consumed by WMMA, then ds_stored to the alternate
// buffer. One barrier per K-step. Guards are branchless (clamp + select).
#define GBM 64
#define GBN 32
#define GBK 32

__launch_bounds__(256)
__global__ void gemm_kernel(const float* __restrict__ A, int lda,
                            const float* __restrict__ B, int ldb,
                            const float* __restrict__ bias,
                            float* __restrict__ C, int ldc,
                            int M, int N, int K, int act, int ntA)
{
    __shared__ __align__(16) float As[2][GBM][GBK];   // 2 x 8 KB
    __shared__ __align__(16) float Bs[2][GBK][GBN];   // 2 x 4 KB

    const int tid  = threadIdx.x;
    const int lane = tid & 31;
    const int wave = tid >> 5;      // 0..7
    const int wm   = wave >> 1;     // 0..3
    const int wn   = wave & 1;      // 0..1
    const int bm0  = blockIdx.y * GBM;
    const int bn0  = blockIdx.x * GBN;

    // per-thread staging coordinates (fixed for the whole kernel)
    const int ar  = tid >> 2;            // A tile row 0..63
    const int ac0 = (tid & 3) << 3;      // A col base {0,8,16,24}
    const int br  = tid >> 3;            // B tile k-row 0..31
    const int bc0 = (tid & 7) << 2;      // B col base {0,4,...,28}

    const int  garow = bm0 + ar;
    const bool aokr  = garow < M;
    const float* __restrict__ Arow = A + (size_t)(aokr ? garow : 0) * lda;

    float areg[8];
    float breg[4];

    auto load_tile = [&](int k0) {
#pragma unroll
        for (int j = 0; j < 8; ++j) {
            int gk = k0 + ac0 + j;
            bool ok = aokr && (gk < K);
            const float* p = Arow + (ok ? gk : 0);
            float v = ntA ? __builtin_nontemporal_load(p) : *p;
            areg[j] = ok ? v : 0.f;
        }
        int gkb = k0 + br;
        const float* __restrict__ Brow = B + (size_t)(gkb < K ? gkb : 0) * ldb;
#pragma unroll
        for (int j = 0; j < 4; ++j) {
            int gc = bn0 + bc0 + j;
            bool ok = (gkb < K) && (gc < N);
            float v = Brow[ok ? gc : 0];
            breg[j] = ok ? v : 0.f;
        }
    };
    auto store_tile = [&](int buf) {
#pragma unroll
        for (int j = 0; j < 8; ++j) As[buf][ar][ac0 + j] = areg[j];
#pragma unroll
        for (int j = 0; j < 4; ++j) Bs[buf][br][bc0 + j] = breg[j];
    };

    v8f acc = {0.f, 0.f, 0.f, 0.f, 0.f, 0.f, 0.f, 0.f};

    load_tile(0);
    store_tile(0);
    __syncthreads();

    const int ml = wm * 16 + (lane & 15);
    const int nl = wn * 16 + (lane & 15);
    const int kh = (lane >> 4) << 1;   // 0 or 2

    int buf = 0;
    for (int k0 = 0; k0 < K; k0 += GBK) {
        const int k1 = k0 + GBK;
        if (k1 < K) load_tile(k1);     // global loads in flight during WMMA stage
#pragma unroll
        for (int kk = 0; kk < GBK; kk += 4) {
            v2f a, b;
            a.x = As[buf][ml][kk + kh];
            a.y = As[buf][ml][kk + kh + 1];
            b.x = Bs[buf][kk + kh][nl];
            b.y = Bs[buf][kk + kh + 1][nl];
            acc = __builtin_amdgcn_wmma_f32_16x16x4_f32(false, a, false, b,
                                                        (short)0, acc, false, false);
        }
        if (k1 < K) store_tile(buf ^ 1);
        __syncthreads();
        buf ^= 1;
    }

    const int col = bn0 + wn * 16 + (lane & 15);
    if (col < N) {
        const float bv = bias ? bias[col] : 0.f;
        const int rbase = bm0 + wm * 16 + ((lane >> 4) << 3);
#pragma unroll
        for (int r = 0; r < 8; ++r) {
            int row = rbase + r;
            if (row < M) {
                float v = acc[r] + bv;
                if (act == 1)      v = (v >= 0.f) ? v : 0.01f * v;           // LeakyReLU
                else if (act == 2) v = 0.5f * v * (1.f + erff(v * 0.7071067811865476f)); // exact GELU
                C[(size_t)row * ldc + col] = v;
            }
        }
    }
}

// ---------------- edge / softmax kernels ----------------
__global__ void softmax_init_kernel(float* m, float* den, float* agg, int Nn)
{
    int i = blockIdx.x * blockDim.x + threadIdx.x;
    if (i < Nn) { m[i] = -INFINITY; den[i] = 0.f; }
    if (i < Nn * HIDC) agg[i] = 0.f;
}

// one wave32 per edge: alpha = dot(q[dst], k_rel[etype][src]) * p[etype] / sqrt(HID)
__global__ void edge_alpha_kernel(const float* __restrict__ kqv,
                                  const float* __restrict__ ktb,
                                  const int* __restrict__ src,
                                  const int* __restrict__ dst,
                                  const int* __restrict__ et,
                                  const float* __restrict__ p,
                                  float* __restrict__ alpha, int E)
{
    int e = blockIdx.x * 8 + (threadIdx.x >> 5);
    int lane = threadIdx.x & 31;
    if (e >= E) return;
    int s = src[e], d = dst[e], t = et[e];
    const float* q  = kqv + (size_t)d * (3 * HIDC) + HIDC;     // q split
    const float* kk = ktb + (size_t)s * (2 * HIDC) + t * HIDC;
    float acc = 0.f;
#pragma unroll
    for (int j = 0; j < HIDC / 32; ++j) {
        int f = lane + 32 * j;
        acc += q[f] * kk[f];
    }
#pragma unroll
    for (int o = 16; o > 0; o >>= 1) acc += __shfl_xor(acc, o, 32);
    if (lane == 0) alpha[e] = acc * p[t] * 0.07905694150420949f; // 1/sqrt(160)
}

__device__ inline void atomicMaxF(float* addr, float val)
{
    unsigned int* ua = (unsigned int*)addr;
    unsigned int cur = *ua;
    while (__uint_as_float(cur) < val) {
        unsigned int old = atomicCAS(ua, cur, __float_as_uint(val));
        if (old == cur) break;
        cur = old;
    }
}

__global__ void edge_max_kernel(const float* __restrict__ alpha,
                                const int* __restrict__ dst,
                                float* m, int E)
{
    int e = blockIdx.x * blockDim.x + threadIdx.x;
    if (e >= E) return;
    atomicMaxF(&m[dst[e]], alpha[e]);
}

__global__ void edge_expw_kernel(float* alpha, const int* __restrict__ dst,
                                 const float* __restrict__ m, float* den, int E)
{
    int e = blockIdx.x * blockDim.x + threadIdx.x;
    if (e >= E) return;
    int d = dst[e];
    float w = expf(alpha[e] - m[d]);
    alpha[e] = w;
    atomicAdd(&den[d], w);
}

// one wave32 per edge: agg[dst] += v_rel[etype][src] * (w / den[dst])
__global__ void edge_agg_kernel(const float* __restrict__ vtb,
                                const float* __restrict__ alpha,
                                const float* __restrict__ den,
                                const int* __restrict__ src,
                                const int* __restrict__ dst,
                                const int* __restrict__ et,
                                float* agg, int E)
{
    int e = blockIdx.x * 8 + (threadIdx.x >> 5);
    int lane = threadIdx.x & 31;
    if (e >= E) return;
    int s = src[e], d = dst[e], t = et[e];
    float a = alpha[e] / fmaxf(den[d], 1e-16f);
    const float* v = vtb + (size_t)s * (2 * HIDC) + t * HIDC;
    float* o = agg + (size_t)d * HIDC;
#pragma unroll
    for (int j = 0; j < HIDC / 32; ++j) {
        int f = lane + 32 * j;
        atomicAdd(&o[f], v[f] * a);
    }
}

__global__ void gelu_kernel(float* x, int n)
{
    int i = blockIdx.x * blockDim.x + threadIdx.x;
    if (i >= n) return;
    float v = x[i];
    x[i] = 0.5f * v * (1.f + erff(v * 0.7071067811865476f));
}

__global__ void blend_kernel(float* h, const float* __restrict__ out,
                             const float* __restrict__ skip, int n)
{
    int i = blockIdx.x * blockDim.x + threadIdx.x;
    if (i >= n) return;
    float g = 1.f / (1.f + expf(-skip[0]));
    h[i] = g * out[i] + (1.f - g) * h[i];
}

// final 80->2 head, one thread per node
__global__ void head_kernel(const float* __restrict__ em,
                            const float* __restrict__ W,
                            const float* __restrict__ b,
                            float* __restrict__ out, int Nn)
{
    int n = blockIdx.x * blockDim.x + threadIdx.x;
    if (n >= Nn) return;
    float a0 = b[0], a1 = b[1];
    const float* e = em + (size_t)n * 80;
#pragma unroll 4
    for (int k = 0; k < 80; ++k) {
        float x = e[k];
        a0 += x * W[k * 2 + 0];
        a1 += x * W[k * 2 + 1];
    }
    out[(size_t)n * 2 + 0] = a0;
    out[(size_t)n * 2 + 1] = a1;
}

// ---------------- launch ----------------
extern "C" void kernel_launch(void* const* d_in, const int* in_sizes, int n_in,
                              void* d_out, int out_size, void* d_ws, size_t ws_size,
                              hipStream_t stream)
{
    const float* pre_x   = (const float*)d_in[0];
    const float* num_prop= (const float*)d_in[2];
    const float* num_cat = (const float*)d_in[3];
    const float* des     = (const float*)d_in[4];
    const float* tweet   = (const float*)d_in[5];
    const int*   eidx    = (const int*)d_in[6];
    const int*   etype   = (const int*)d_in[7];
    const float* W_np = (const float*)d_in[8],  *b_np = (const float*)d_in[9];
    const float* W_nc = (const float*)d_in[10], *b_nc = (const float*)d_in[11];
    const float* W_des= (const float*)d_in[12], *b_des= (const float*)d_in[13];
    const float* W_txt= (const float*)d_in[14], *b_txt= (const float*)d_in[15];
    const float* W_tw = (const float*)d_in[16], *b_tw = (const float*)d_in[17];
    const float* W_in = (const float*)d_in[18], *b_in = (const float*)d_in[19];
    const float* W_o1 = (const float*)d_in[36], *b_o1 = (const float*)d_in[37];
    const float* W_o2 = (const float*)d_in[38], *b_o2 = (const float*)d_in[39];

    const int Nn = in_sizes[2] / 6;     // 50000
    const int E  = in_sizes[7];         // 500000
    const int* src = eidx;
    const int* dst = eidx + E;

    // workspace layout (floats)
    float* ws = (float*)d_ws;
    size_t o = 0;
    float* h    = ws + o; o += (size_t)Nn * HIDC;
    float* feat = ws + o; o += (size_t)Nn * HIDC;      // also reused as layer tmp
    float* kqv  = ws + o; o += (size_t)Nn * 3 * HIDC;
    float* ktb  = ws + o; o += (size_t)Nn * 2 * HIDC;
    float* vtb  = ws + o; o += (size_t)Nn * 2 * HIDC;
    float* agg  = ws + o; o += (size_t)Nn * HIDC;
    float* alph = ws + o; o += (size_t)E;
    float* mbuf = ws + o; o += (size_t)Nn;
    float* dbuf = ws + o; o += (size_t)Nn;
    (void)n_in; (void)out_size; (void)ws_size;

    auto gemm = [&](const float* A, int lda, const float* B, int ldb,
                    const float* bias, float* C, int ldc,
                    int M, int N, int K, int act, int ntA) {
        dim3 grid((unsigned)((N + GBN - 1) / GBN), (unsigned)((M + GBM - 1) / GBM));
        gemm_kernel<<<grid, dim3(256), 0, stream>>>(A, lda, B, ldb, bias, C, ldc,
                                                    M, N, K, act, ntA);
    };

    // ---- feature encoders into feat[N,160] (concat order: np, nc, des, tweet@W_txt, pre_x@W_tw)
    // The three 768-wide streams are read exactly once -> non-temporal A loads.
    gemm(num_prop, 6,   W_np, 32, b_np, feat + 0,   HIDC, Nn, 32, 6,   1, 0);
    gemm(num_cat, 11,   W_nc, 32, b_nc, feat + 32,  HIDC, Nn, 32, 11,  1, 0);
    gemm(des,     768,  W_des,32, b_des,feat + 64,  HIDC, Nn, 32, 768, 1, 1);
    gemm(tweet,   768,  W_txt,32, b_txt,feat + 96,  HIDC, Nn, 32, 768, 1, 1);
    gemm(pre_x,   768,  W_tw, 32, b_tw, feat + 128, HIDC, Nn, 32, 768, 1, 1);
    // ---- input projection
    gemm(feat, HIDC, W_in, HIDC, b_in, h, HIDC, Nn, HIDC, HIDC, 1, 0);

    const int nelem = Nn * HIDC;
    const unsigned gElem  = (unsigned)((nelem + 255) / 256);
    const unsigned gEdgeW = (unsigned)((E + 7) / 8);      // wave-per-edge
    const unsigned gEdgeT = (unsigned)((E + 255) / 256);  // thread-per-edge

    // ---- two HGT layers
    for (int l = 0; l < 2; ++l) {
        const float* Wkqv = (const float*)d_in[20 + 8 * l];
        const float* bkqv = (const float*)d_in[21 + 8 * l];
        const float* kW   = (const float*)d_in[22 + 8 * l];  // [2,160,160]
        const float* vW   = (const float*)d_in[23 + 8 * l];
        const float* prel = (const float*)d_in[24 + 8 * l];  // [2]
        const float* skip = (const float*)d_in[25 + 8 * l];  // scalar
        const float* Wout = (const float*)d_in[26 + 8 * l];
        const float* bout = (const float*)d_in[27 + 8 * l];

        // kqv = h @ Wkqv + b   [N,480]; splits: k=0..159, q=160..319, v=320..479
        gemm(h, HIDC, Wkqv, 3 * HIDC, bkqv, kqv, 3 * HIDC, Nn, 3 * HIDC, HIDC, 0, 0);
        // per-edge-type relation transforms
        for (int t = 0; t < 2; ++t) {
            gemm(kqv + 0,        3 * HIDC, kW + (size_t)t * HIDC * HIDC, HIDC, nullptr,
                 ktb + t * HIDC, 2 * HIDC, Nn, HIDC, HIDC, 0, 0);
            gemm(kqv + 2 * HIDC, 3 * HIDC, vW + (size_t)t * HIDC * HIDC, HIDC, nullptr,
                 vtb + t * HIDC, 2 * HIDC, Nn, HIDC, HIDC, 0, 0);
        }

        softmax_init_kernel<<<gElem, 256, 0, stream>>>(mbuf, dbuf, agg, Nn);
        edge_alpha_kernel<<<gEdgeW, 256, 0, stream>>>(kqv, ktb, src, dst, etype, prel, alph, E);
        edge_max_kernel <<<gEdgeT, 256, 0, stream>>>(alph, dst, mbuf, E);
        edge_expw_kernel<<<gEdgeT, 256, 0, stream>>>(alph, dst, mbuf, dbuf, E);
        edge_agg_kernel <<<gEdgeW, 256, 0, stream>>>(vtb, alph, dbuf, src, dst, etype, agg, E);

        gelu_kernel<<<gElem, 256, 0, stream>>>(agg, nelem);
        // out = gelu(agg) @ Wout + bout  -> feat (tmp)
        gemm(agg, HIDC, Wout, HIDC, bout, feat, HIDC, Nn, HIDC, HIDC, 0, 0);
        // h = g*out + (1-g)*h
        blend_kernel<<<gElem, 256, 0, stream>>>(h, feat, skip, nelem);
    }

    // ---- head: em = leaky(h @ W_o1 + b_o1) -> d_out[2N .. 2N+80N)
    float* out = (float*)d_out;
    float* em  = out + (size_t)2 * Nn;
    gemm(h, HIDC, W_o1, 80, b_o1, em, 80, Nn, 80, HIDC, 1, 0);
    // out = em @ W_o2 + b_o2 -> d_out[0 .. 2N)
    head_kernel<<<(unsigned)((Nn + 255) / 256), 256, 0, stream>>>(em, W_o2, b_o2, out, Nn);
}